// DumberDecoder_8272107012806
// MI455X (gfx1250) — compile-verified
//
#include <hip/hip_runtime.h>
#include <math.h>

#define ALPH 1024
#define TAG_ 512
#define EMB_ 512
#define CTX_ 256
#define ABS_ 60
#define ATT_ 100
#define SSEQ 2048
#define TOUT 2048
#define DKIN (ABS_ + CTX_ + EMB_)                    // 828
#define DFF  ((CTX_+EMB_) + (CTX_+ALPH) + CTX_ + ALPH + TAG_)  // 3840
#define NSTEP (TOUT - 2)                             // 2046 steps: t = 1..2046
#define NWG 32
#define SA (SSEQ / NWG)                              // 64 attention rows / WG
#define SP (ALPH / NWG)                              // 32 ff outputs / WG
#define SW (CTX_ / NWG)                              // 8 win outputs / WG
#define NPAD 112                                     // 7 N-tiles of 16

typedef __attribute__((ext_vector_type(2))) float v2f;
typedef __attribute__((ext_vector_type(8))) float v8f;

__device__ __forceinline__ float sigf(float x) { return 1.0f / (1.0f + __expf(-x)); }

// ---------------- workspace layout (floats) ----------------
constexpr size_t OFF_KIN   = 0;                               // 2048 x 828
constexpr size_t SZ_KIN    = (size_t)SSEQ * DKIN;
constexpr size_t OFF_KEYS  = OFF_KIN + SZ_KIN;                // 2048 x 100
constexpr size_t OFF_POSQ  = OFF_KEYS + (size_t)SSEQ * ATT_;  // 2048 x 100
constexpr size_t OFF_QC    = OFF_POSQ + (size_t)SSEQ * ATT_;  // 128 (pad)
constexpr size_t OFF_FFC   = OFF_QC + 128;                    // 1024
constexpr size_t OFF_PBUF  = OFF_FFC + 1024;                  // 2 x 1024
constexpr size_t OFF_WBUF  = OFF_PBUF + 2 * ALPH;             // 2 x 256
constexpr size_t OFF_APART = OFF_WBUF + 2 * CTX_;             // NWG x 768
constexpr size_t OFF_RMAX  = OFF_APART + (size_t)NWG * 768;   // NWG
constexpr size_t OFF_RSUM  = OFF_RMAX + NWG;                  // NWG
constexpr size_t OFF_RPSUM = OFF_RSUM + NWG;                  // NWG
constexpr size_t OFF_PTMP  = OFF_RPSUM + NWG;                 // 1024
constexpr size_t OFF_BKT   = OFF_PTMP + ALPH;                 // 112 x 828 (Wk^T padded)
constexpr size_t OFF_BQT   = OFF_BKT + (size_t)NPAD * DKIN;   // 112 x 60  (Wq_pos^T padded)
constexpr size_t OFF_BAR   = OFF_BQT + (size_t)NPAD * ABS_;   // 2 uints

// ---------------- build Kin = [sinusoidal pos | states | embedding] ----------------
__global__ void build_kin_kernel(const float* __restrict__ states,
                                 const float* __restrict__ emb,
                                 float* __restrict__ kin) {
  size_t idx = (size_t)blockIdx.x * blockDim.x + threadIdx.x;
  if (idx >= (size_t)SSEQ * DKIN) return;
  int row = (int)(idx / DKIN), c = (int)(idx % DKIN);
  float v;
  if (c < ABS_) {
    int i = c >> 1;
    float ang = (float)row * powf(10000.0f, -2.0f * (float)i / (float)ABS_);
    v = (c & 1) ? cosf(ang) : sinf(ang);
  } else if (c < ABS_ + CTX_) {
    v = states[(size_t)row * CTX_ + (c - ABS_)];
  } else {
    v = emb[(size_t)row * EMB_ + (c - ABS_ - CTX_)];
  }
  kin[idx] = v;
}

// ---------------- pad+transpose B (K x N, ld=N) -> BT (NPAD x K), zero pad cols ----------
__global__ void pad_bt_kernel(const float* __restrict__ B, int K, int N,
                              float* __restrict__ BT) {
  size_t idx = (size_t)blockIdx.x * blockDim.x + threadIdx.x;
  if (idx >= (size_t)NPAD * K) return;
  int c = (int)(idx / K), k = (int)(idx % K);
  BT[idx] = (c < N) ? B[(size_t)k * N + c] : 0.f;
}

// ---------------- WMMA fp32 GEMM: C(2048 x N) = A(2048 x K) @ B(K x N) [+bias][sigmoid] ----
// B is pre-transposed+padded: BT[NPAD][K]. One 16x16 tile per wave, K stepped by 4 with
// V_WMMA_F32_16X16X4_F32; branchless inner loop = 2x global_load_b64 + 1x wmma.
// A layout (16x4 f32): lanes0-15 M=0..15; VGPR0=K0|K2, VGPR1=K1|K3.
// B layout (4x16 f32): VGPR0 = row K0 (lanes0-15) | row K2 (lanes16-31); VGPR1 = K1|K3.
// C/D (16x16 f32): VGPR i = row i (lanes0-15) | row i+8 (lanes16-31), N = lane&15.
__global__ void __launch_bounds__(128, 1)
gemm16_wmma_kernel(const float* __restrict__ A, int lda,
                   const float* __restrict__ BT,
                   int K, int N, const float* __restrict__ bias, int act,
                   float* __restrict__ C, int ldc, int ntileN) {
  int wave = threadIdx.x >> 5;
  int lane = threadIdx.x & 31;
  int tile = blockIdx.x * 4 + wave;
  int m0 = (tile / ntileN) * 16;
  int n0 = (tile % ntileN) * 16;
  int half = lane >> 4, rr = lane & 15;
  int col = n0 + rr;                       // < NPAD always; zero-padded beyond N
  v8f acc = {0.f, 0.f, 0.f, 0.f, 0.f, 0.f, 0.f, 0.f};
  const float* arow = A + (size_t)(m0 + rr) * lda + 2 * half;   // 8B aligned
  const float* bcol = BT + (size_t)col * K + 2 * half;          // 8B aligned
  for (int k = 0; k < K; k += 4) {
    v2f a = *(const v2f*)(arow + k);       // A[m][k+2h], A[m][k+2h+1]
    v2f b = *(const v2f*)(bcol + k);       // B[k+2h][col], B[k+2h+1][col]
    acc = __builtin_amdgcn_wmma_f32_16x16x4_f32(false, a, false, b, (short)0, acc,
                                                false, false);
  }
  if (col < N) {
    float bb = bias ? bias[col] : 0.f;
#pragma unroll
    for (int i = 0; i < 8; ++i) {
      int row = m0 + i + 8 * half;
      float v = acc[i] + bb;
      if (act) v = sigf(v);
      C[(size_t)row * ldc + col] = v;
    }
  }
}

// ---------------- fold tags/bias into constant vectors ----------------
__global__ void consts_kernel(const float* __restrict__ tags,
                              const float* __restrict__ Wq, const float* __restrict__ bq,
                              const float* __restrict__ Wff, const float* __restrict__ bff,
                              float* __restrict__ qc, float* __restrict__ ffc) {
  int tid = threadIdx.x;
  if (tid < ATT_) {
    float acc = bq[tid];
    for (int i = 0; i < TAG_; ++i)
      acc += tags[i] * Wq[(size_t)(ABS_ + CTX_ + ALPH + i) * ATT_ + tid];
    qc[tid] = acc;
  }
  for (int j = tid; j < ALPH; j += 256) {
    float acc = bff[j];
    for (int i = 0; i < TAG_; ++i)
      acc += tags[i] * Wff[(size_t)(DFF - TAG_ + i) * ALPH + j];
    ffc[j] = acc;
  }
}

// ---------------- init: out row0 = one-hot(START=1), p_buf[1], win_buf[1], barrier --------
__global__ void init_kernel(float* __restrict__ out, float* __restrict__ pbuf,
                            float* __restrict__ wbuf, unsigned* __restrict__ bar) {
  int tid = threadIdx.x;
  for (int i = tid; i < ALPH; i += 256) {
    float v = (i == 1) ? 1.0f : 0.0f;
    out[i] = v;            // output row 0
    pbuf[ALPH + i] = v;    // p_buf[parity=1] = p0
  }
  for (int i = tid; i < CTX_; i += 256) wbuf[CTX_ + i] = 0.f;  // win_buf[1] = 0
  if (tid < 2) bar[tid] = 0u;
}

// ---------------- device-scope barrier (generation counting) ----------------
__device__ __forceinline__ void gbarrier(unsigned* cnt, unsigned* gen, unsigned& epoch) {
  epoch += 1;
  __threadfence();
  __syncthreads();
  if (threadIdx.x == 0) {
    unsigned prev = __hip_atomic_fetch_add(cnt, 1u, __ATOMIC_ACQ_REL,
                                           __HIP_MEMORY_SCOPE_AGENT);
    if (prev == NWG - 1u) {
      __hip_atomic_store(cnt, 0u, __ATOMIC_RELAXED, __HIP_MEMORY_SCOPE_AGENT);
      __hip_atomic_fetch_add(gen, 1u, __ATOMIC_RELEASE, __HIP_MEMORY_SCOPE_AGENT);
    } else {
      while (__hip_atomic_load(gen, __ATOMIC_ACQUIRE, __HIP_MEMORY_SCOPE_AGENT) < epoch) {
        __builtin_amdgcn_s_sleep(1);
      }
    }
  }
  __syncthreads();
  __threadfence();
}

// ---------------- persistent sequential decoder: NWG=32 WGs x 256 threads ----------------
__global__ void __launch_bounds__(256, 1)
decoder_kernel(float* __restrict__ ws,
               const float* __restrict__ W1, const float* __restrict__ b1,
               const float* __restrict__ W2, const float* __restrict__ b2,
               const float* __restrict__ Wq, const float* __restrict__ Wff,
               float* __restrict__ out) {
  const float* kin  = ws + OFF_KIN;
  const float* keys = ws + OFF_KEYS;
  const float* posq = ws + OFF_POSQ;
  const float* qc   = ws + OFF_QC;
  const float* ffc  = ws + OFF_FFC;
  float* pbuf  = ws + OFF_PBUF;
  float* wbuf  = ws + OFF_WBUF;
  float* apart = ws + OFF_APART;
  float* rmax  = ws + OFF_RMAX;
  float* rsum  = ws + OFF_RSUM;
  float* rpsum = ws + OFF_RPSUM;
  float* ptmp  = ws + OFF_PTMP;
  unsigned* bar = (unsigned*)(ws + OFF_BAR);

  __shared__ float pprev_s[ALPH];
  __shared__ float win_s[CTX_];
  __shared__ float ctx_s[CTX_];
  __shared__ float q_s[112];
  __shared__ float e_s[SA];
  __shared__ float attn_s[CTX_ + EMB_];  // 768
  __shared__ float wpart_s[8][32];
  __shared__ float red_s[256];
  __shared__ float scal_s[2];

  const int tid = threadIdx.x;
  const int g = blockIdx.x;
  const int wv = tid >> 5, ln = tid & 31;
  unsigned epoch = 0;

  for (int t = 1; t <= NSTEP; ++t) {
    const int r = t & 1, w = r ^ 1;
    // ---- Phase A: load state, ctx (redundant), win_next (distributed), q (redundant),
    //               logits slice + block max ----
    for (int i = tid; i < ALPH; i += 256) pprev_s[i] = pbuf[(size_t)r * ALPH + i];
    for (int i = tid; i < CTX_; i += 256) win_s[i] = wbuf[(size_t)r * CTX_ + i];
    __syncthreads();
    {  // ctx = sigmoid(win @ W2 + b2), all 256 outputs per WG
      float acc = b2[tid];
      for (int k = 0; k < CTX_; ++k) acc += win_s[k] * W2[(size_t)k * CTX_ + tid];
      ctx_s[tid] = sigf(acc);
    }
    {  // win_next slice: wave wv -> output g*8+wv, lanes split K=1024
      int jw = g * SW + wv;
      float acc = 0.f;
      for (int it = 0; it < 32; ++it) {
        int k = ln + (it << 5);
        acc += pprev_s[k] * W1[(size_t)k * CTX_ + jw];
      }
      wpart_s[wv][ln] = acc;
    }
    __syncthreads();
    if (tid < SW) {
      int jw = g * SW + tid;
      float acc = b1[jw];
      for (int l2 = 0; l2 < 32; ++l2) acc += wpart_s[tid][l2];
      wbuf[(size_t)w * CTX_ + jw] = sigf(acc);
    }
    if (tid < ATT_) {  // q = sigmoid(pos_t@Wq + ctx@Wq + p_prev@Wq + tags@Wq + bq)
      float acc = posq[(size_t)t * ATT_ + tid] + qc[tid];
      for (int k = 0; k < CTX_; ++k)
        acc += ctx_s[k] * Wq[(size_t)(ABS_ + k) * ATT_ + tid];
      for (int k = 0; k < ALPH; ++k)
        acc += pprev_s[k] * Wq[(size_t)(ABS_ + CTX_ + k) * ATT_ + tid];
      q_s[tid] = sigf(acc);
    }
    __syncthreads();
    float lval = -3.0e38f;
    if (tid < SA) {
      int srow = g * SA + tid;
      float acc = 0.f;
      for (int k = 0; k < ATT_; ++k) acc += keys[(size_t)srow * ATT_ + k] * q_s[k];
      lval = acc;
    }
    red_s[tid] = lval;
    __syncthreads();
    for (int s2 = 128; s2 > 0; s2 >>= 1) {
      if (tid < s2) red_s[tid] = fmaxf(red_s[tid], red_s[tid + s2]);
      __syncthreads();
    }
    if (tid == 0) rmax[g] = red_s[0];
    gbarrier(bar, bar + 1, epoch);  // A

    // ---- Phase B: global max, e = exp(l - max), partial Z, partial attn ----
    if (tid == 0) {
      float m = -3.0e38f;
      for (int i = 0; i < NWG; ++i) m = fmaxf(m, rmax[i]);
      scal_s[0] = m;
    }
    __syncthreads();
    float ev = 0.f;
    if (tid < SA) {
      ev = __expf(lval - scal_s[0]);
      e_s[tid] = ev;
    }
    red_s[tid] = ev;
    __syncthreads();
    for (int s2 = 128; s2 > 0; s2 >>= 1) {
      if (tid < s2) red_s[tid] += red_s[tid + s2];
      __syncthreads();
    }
    if (tid == 0) rsum[g] = red_s[0];
#pragma unroll
    for (int c = 0; c < 3; ++c) {  // partial attn over this WG's s-slice, all 768 cols
      int j = tid + 256 * c;
      const float* vbase = kin + (size_t)(g * SA) * DKIN + ABS_ + j;
      float acc = 0.f;
      for (int s3 = 0; s3 < SA; ++s3) acc += e_s[s3] * vbase[(size_t)s3 * DKIN];
      apart[(size_t)g * 768 + j] = acc;
    }
    gbarrier(bar, bar + 1, epoch);  // B

    // ---- Phase C: reduce attn (redundant), ff slice + sigmoid + exp + partial sum ----
    if (tid == 0) {
      float z = 0.f;
      for (int i = 0; i < NWG; ++i) z += rsum[i];
      scal_s[1] = 1.0f / z;
    }
    __syncthreads();
#pragma unroll
    for (int c = 0; c < 3; ++c) {
      int j = tid + 256 * c;
      float a = 0.f;
      for (int gg = 0; gg < NWG; ++gg) a += apart[(size_t)gg * 768 + j];
      attn_s[j] = a * scal_s[1];
    }
    __syncthreads();
    {  // ff: wave wv handles a 256-row band of Wff for this WG's 32 output columns
      int j = g * SP + ln;
      const float* src;
      int row0;
      if (wv < 3)       { src = attn_s + wv * 256;        row0 = wv * 256; }
      else if (wv == 3) { src = ctx_s;                    row0 = 2048; }
      else              { src = pprev_s + (wv - 4) * 256; row0 = 2304 + (wv - 4) * 256; }
      float acc = 0.f;
      for (int kk = 0; kk < 256; ++kk)
        acc += src[kk] * Wff[(size_t)(row0 + kk) * ALPH + j];
      wpart_s[wv][ln] = acc;
    }
    __syncthreads();
    if (tid < 32) {
      int j = g * SP + tid;
      float tot = ffc[j];
      for (int w2 = 0; w2 < 8; ++w2) tot += wpart_s[w2][tid];
      float ep = __expf(sigf(tot));  // sigmoid in (0,1) -> exp safe without max-shift
      ptmp[j] = ep;
      red_s[tid] = ep;
    }
    __syncthreads();
    if (tid == 0) {
      float z = 0.f;
      for (int i = 0; i < 32; ++i) z += red_s[i];
      rpsum[g] = z;
    }
    gbarrier(bar, bar + 1, epoch);  // C

    // ---- Phase D: normalize p_new, write state + output ----
    if (tid == 0) {
      float z = 0.f;
      for (int i = 0; i < NWG; ++i) z += rpsum[i];
      scal_s[0] = 1.0f / z;
    }
    __syncthreads();
    if (tid < SP) {
      int j = g * SP + tid;
      float v = ptmp[j] * scal_s[0];
      pbuf[(size_t)w * ALPH + j] = v;
      out[(size_t)t * ALPH + j] = v;
    }
    gbarrier(bar, bar + 1, epoch);  // D (loop-carried dependence)
  }
}

// ---------------- host launcher ----------------
extern "C" void kernel_launch(void* const* d_in, const int* in_sizes, int n_in,
                              void* d_out, int out_size, void* d_ws, size_t ws_size,
                              hipStream_t stream) {
  const float* tags  = (const float*)d_in[2];
  const float* emb   = (const float*)d_in[3];
  const float* states= (const float*)d_in[4];
  const float* W1  = (const float*)d_in[6];
  const float* b1  = (const float*)d_in[7];
  const float* W2  = (const float*)d_in[8];
  const float* b2  = (const float*)d_in[9];
  const float* Wq  = (const float*)d_in[10];
  const float* bq  = (const float*)d_in[11];
  const float* Wk  = (const float*)d_in[12];
  const float* bk  = (const float*)d_in[13];
  const float* Wff = (const float*)d_in[14];
  const float* bff = (const float*)d_in[15];
  float* out = (float*)d_out;
  float* ws  = (float*)d_ws;

  // init (barrier state, ping-pong state, output row 0)
  init_kernel<<<1, 256, 0, stream>>>(out, ws + OFF_PBUF, ws + OFF_WBUF,
                                     (unsigned*)(ws + OFF_BAR));
  // Kin = [pos | states | embedding]
  {
    size_t n = (size_t)SSEQ * DKIN;
    build_kin_kernel<<<(unsigned)((n + 255) / 256), 256, 0, stream>>>(states, emb,
                                                                      ws + OFF_KIN);
  }
  // padded/transposed B operands for branchless WMMA inner loops
  {
    size_t n1 = (size_t)NPAD * DKIN;   // Wk^T padded (112 x 828)
    pad_bt_kernel<<<(unsigned)((n1 + 255) / 256), 256, 0, stream>>>(Wk, DKIN, ATT_,
                                                                    ws + OFF_BKT);
    size_t n2 = (size_t)NPAD * ABS_;   // Wq_pos^T padded (112 x 60)
    pad_bt_kernel<<<(unsigned)((n2 + 255) / 256), 256, 0, stream>>>(Wq, ABS_, ATT_,
                                                                    ws + OFF_BQT);
  }
  // keys = sigmoid(Kin @ Wk + bk)   : 2048x828x100 fp32 WMMA GEMM
  gemm16_wmma_kernel<<<32 * 7, 128, 0, stream>>>(ws + OFF_KIN, DKIN, ws + OFF_BKT,
                                                 DKIN, ATT_, bk, 1,
                                                 ws + OFF_KEYS, ATT_, 7);
  // posq = pos @ Wq[0:60,:]         : 2048x60x100 fp32 WMMA GEMM (no bias/act)
  gemm16_wmma_kernel<<<32 * 7, 128, 0, stream>>>(ws + OFF_KIN, DKIN, ws + OFF_BQT,
                                                 ABS_, ATT_, nullptr, 0,
                                                 ws + OFF_POSQ, ATT_, 7);
  // fold tags + biases into constants
  consts_kernel<<<1, 256, 0, stream>>>(tags, Wq, bq, Wff, bff,
                                       ws + OFF_QC, ws + OFF_FFC);
  // persistent sequential decoder
  decoder_kernel<<<NWG, 256, 0, stream>>>(ws, W1, b1, W2, b2, Wq, Wff, out);
}